// ModulatedPlasticDense_403726926394
// MI455X (gfx1250) — compile-verified
//
#include <hip/hip_runtime.h>

#define B_    512
#define IN_   1023
#define K_    1024   // IN+1 (bias column of ones)
#define OUT_  512
#define CLIPV 2.0f
#define LNEPS 1e-5f

typedef __attribute__((ext_vector_type(2))) float f32x2;
typedef __attribute__((ext_vector_type(4))) float f32x4;
typedef __attribute__((ext_vector_type(8))) float f32x8;

// ---------------------------------------------------------------------------
// Kernel 1: base[b,o] = xp[b,:] @ weight[:,o]  via V_WMMA_F32_16X16X4_F32.
// One wave per 16x16 output tile. A is 16x4 f32 (2 VGPRs/lane), B is 4x16
// (2 VGPRs/lane), C/D 16x16 f32 (8 VGPRs/lane).
//   A layout: lane L (L<16): m=L, V0=K0, V1=K1 ; lane>=16: m=L-16, V0=K2, V1=K3
//   B layout (mirrored): lane L: n=L%16, V0=K(2*half), V1=K(2*half+1)
//   C layout: lane L: n=L%16, VGPR r -> m = r + 8*(L/16)
// Result is written to the y-region of d_out; kernel 2 consumes it as "base"
// and overwrites it with the final y.
// ---------------------------------------------------------------------------
__global__ __launch_bounds__(128) void mpd_base_gemm_wmma(
    const float* __restrict__ x,        // [B, IN]
    const float* __restrict__ weight,   // [K, OUT]
    float* __restrict__ base)           // [B, OUT]
{
  const int wave = threadIdx.x >> 5;
  const int lane = threadIdx.x & 31;
  const int tile = blockIdx.x * 4 + wave;   // 0..1023 tiles
  const int bt   = tile >> 5;               // batch tile  (B/16  = 32)
  const int ot   = tile & 31;               // out tile    (OUT/16 = 32)
  const int half = lane >> 4;               // 0 | 1
  const int mn   = lane & 15;               // A row / B,C col within tile

  const float* xrow = x + (bt * 16 + mn) * IN_;
  const int    col  = ot * 16 + mn;

  f32x8 c = {};
  for (int k0 = 0; k0 < K_; k0 += 4) {
    const int ka = k0 + 2 * half;
    f32x2 a, bf;
    a.x  = (ka     < IN_) ? xrow[ka]     : 1.0f;   // bias column -> 1.0
    a.y  = (ka + 1 < IN_) ? xrow[ka + 1] : 1.0f;
    bf.x = weight[(ka    ) * OUT_ + col];
    bf.y = weight[(ka + 1) * OUT_ + col];
    // (neg_a, A, neg_b, B, c_mod, C, reuse_a, reuse_b)
    c = __builtin_amdgcn_wmma_f32_16x16x4_f32(false, a, false, bf,
                                              (short)0, c, false, false);
  }
  #pragma unroll
  for (int r = 0; r < 8; ++r) {
    const int m = r + 8 * half;
    base[(bt * 16 + m) * OUT_ + ot * 16 + mn] = c[r];
  }
}

// ---------------------------------------------------------------------------
// Workgroup (256 threads) sum reduction via LDS. All threads must call.
// ---------------------------------------------------------------------------
__device__ __forceinline__ float block_reduce(float v, float* buf, int tid) {
  buf[tid] = v;
  __syncthreads();
  #pragma unroll
  for (int s = 128; s > 0; s >>= 1) {
    if (tid < s) buf[tid] += buf[tid + s];
    __syncthreads();
  }
  float r = buf[0];
  __syncthreads();
  return r;
}

// ---------------------------------------------------------------------------
// Kernel 2: fused plastic term + LayerNorm + tanh + modulator + Hebb update.
// One 256-thread workgroup per sample b.
//   group g = tid>>7 owns i-range [g*512, g*512+512)
//   lane  u = tid&127 owns 4 contiguous output columns o0 = 4u  (b128 traffic)
// Pass 1 streams hebb[b] (2 MB) with RT hints (keep in L2 for pass 2);
// partial accumulators are combined across the two i-groups through LDS.
// Pass 2 re-reads hebb[b] (last use -> NT loads) and writes hebb_new with
// NT stores so dirty lines don't pollute L2.
// ---------------------------------------------------------------------------
__global__ __launch_bounds__(256) void mpd_fused(
    const float* __restrict__ x,         // [B, IN]
    const float* __restrict__ hebb,      // [B, K, OUT]
    const float* __restrict__ alpha,     // [K, OUT]
    const float* __restrict__ ln_gamma,  // [OUT]
    const float* __restrict__ ln_beta,   // [OUT]
    const float* __restrict__ mod_w,     // [1, OUT]
    const float* __restrict__ mod_b,     // [1]
    const float* __restrict__ fan_w,     // [OUT, 1]
    const float* __restrict__ fan_b,     // [OUT]
    float* __restrict__ y_out,           // [B, OUT]  (holds base on entry)
    float* __restrict__ m_out,           // [B]
    float* __restrict__ hebb_out)        // [B, K, OUT]
{
  __shared__ float xp[K_];
  __shared__ float red[256];
  __shared__ float comb[OUT_];   // acc combine, then c[o] broadcast

  const int b   = blockIdx.x;
  const int tid = threadIdx.x;
  const int g   = tid >> 7;      // i-range group: 0 or 1
  const int u   = tid & 127;     // column owner within group
  const int o0  = u * 4;

  // Stage xp[b,:] (with bias 1.0) in LDS for broadcast reads.
  for (int i = tid; i < K_; i += 256)
    xp[i] = (i < IN_) ? x[b * IN_ + i] : 1.0f;
  __syncthreads();

  const f32x4* __restrict__ hb = (const f32x4*)(hebb + (size_t)b * K_ * OUT_);
  const f32x4* __restrict__ al = (const f32x4*)alpha;

  // Pass 1: acc[o] = sum_{i in my half} xp[i]*alpha[i,o]*hebb[b,i,o]
  const int ibeg = g * (K_ / 2);
  float acc[4] = {0.f, 0.f, 0.f, 0.f};
  #pragma unroll 4
  for (int ii = 0; ii < K_ / 2; ++ii) {
    const int   i  = ibeg + ii;
    const float xi = xp[i];
    const f32x4 h  = hb[i * (OUT_ / 4) + u];
    const f32x4 a  = al[i * (OUT_ / 4) + u];
    #pragma unroll
    for (int j = 0; j < 4; ++j)
      acc[j] = fmaf(xi * a[j], h[j], acc[j]);
  }

  // Combine the two i-halves: group1 publishes, group0 accumulates (+base).
  if (g == 1) {
    #pragma unroll
    for (int j = 0; j < 4; ++j) comb[o0 + j] = acc[j];
  }
  __syncthreads();

  float yp[4] = {0.f, 0.f, 0.f, 0.f};
  if (g == 0) {
    #pragma unroll
    for (int j = 0; j < 4; ++j)
      yp[j] = acc[j] + comb[o0 + j] + y_out[b * OUT_ + o0 + j];  // + base GEMM
  }

  // LayerNorm over OUT (group1 contributes zeros to the reductions).
  const float mu = block_reduce(yp[0] + yp[1] + yp[2] + yp[3], red, tid)
                   * (1.0f / OUT_);
  float d[4] = {0.f, 0.f, 0.f, 0.f};
  float vsum = 0.f;
  if (g == 0) {
    #pragma unroll
    for (int j = 0; j < 4; ++j) { d[j] = yp[j] - mu; vsum += d[j] * d[j]; }
  }
  const float var  = block_reduce(vsum, red, tid) * (1.0f / OUT_);
  const float rstd = rsqrtf(var + LNEPS);

  float yv[4] = {0.f, 0.f, 0.f, 0.f};
  float msum = 0.f;
  if (g == 0) {
    #pragma unroll
    for (int j = 0; j < 4; ++j) {
      yv[j] = tanhf(ln_gamma[o0 + j] * d[j] * rstd + ln_beta[o0 + j]);
      msum += yv[j] * mod_w[o0 + j];
    }
  }

  // Modulator scalar m = tanh(y . mod_w + mod_b)
  const float m = tanhf(block_reduce(msum, red, tid) + mod_b[0]);
  if (tid == 0) m_out[b] = m;

  // Group0 emits y (overwrites base) and publishes per-column coefficients
  // c[o] = (m*fan_w[o]+fan_b[o])*y[o]; delta[i,o] = c[o]*xp[i].
  if (g == 0) {
    f32x4 yo;
    #pragma unroll
    for (int j = 0; j < 4; ++j) yo[j] = yv[j];
    *(f32x4*)(y_out + b * OUT_ + o0) = yo;
    #pragma unroll
    for (int j = 0; j < 4; ++j)
      comb[o0 + j] = (m * fan_w[o0 + j] + fan_b[o0 + j]) * yv[j];
  }
  __syncthreads();

  float c[4];
  #pragma unroll
  for (int j = 0; j < 4; ++j) c[j] = comb[o0 + j];

  // Pass 2: hebb_new = clip(hebb + c[o]*xp[i], +-2). Last-use NT read,
  // NT write (streaming, zero reuse).
  f32x4* __restrict__ ho = (f32x4*)(hebb_out + (size_t)b * K_ * OUT_);
  #pragma unroll 4
  for (int ii = 0; ii < K_ / 2; ++ii) {
    const int   i  = ibeg + ii;
    const float xi = xp[i];
    const f32x4 h  = __builtin_nontemporal_load(&hb[i * (OUT_ / 4) + u]);
    f32x4 o;
    #pragma unroll
    for (int j = 0; j < 4; ++j)
      o[j] = fminf(fmaxf(fmaf(c[j], xi, h[j]), -CLIPV), CLIPV);
    __builtin_nontemporal_store(o, &ho[i * (OUT_ / 4) + u]);
  }
}

extern "C" void kernel_launch(void* const* d_in, const int* in_sizes, int n_in,
                              void* d_out, int out_size, void* d_ws, size_t ws_size,
                              hipStream_t stream) {
  (void)in_sizes; (void)n_in; (void)out_size; (void)d_ws; (void)ws_size;
  const float* x        = (const float*)d_in[0];
  const float* hebb     = (const float*)d_in[1];
  const float* weight   = (const float*)d_in[2];
  const float* alpha    = (const float*)d_in[3];
  const float* ln_gamma = (const float*)d_in[4];
  const float* ln_beta  = (const float*)d_in[5];
  const float* mod_w    = (const float*)d_in[6];
  const float* mod_b    = (const float*)d_in[7];
  const float* fan_w    = (const float*)d_in[8];
  const float* fan_b    = (const float*)d_in[9];

  float* y_out    = (float*)d_out;              // [B, OUT]
  float* m_out    = y_out + (size_t)B_ * OUT_;  // [B]
  float* hebb_out = m_out + B_;                 // [B, K, OUT]

  // 1024 tiles, 4 waves (tiles) per 128-thread block -> 256 blocks.
  mpd_base_gemm_wmma<<<256, 128, 0, stream>>>(x, weight, y_out);
  mpd_fused<<<B_, 256, 0, stream>>>(x, hebb, alpha, ln_gamma, ln_beta,
                                    mod_w, mod_b, fan_w, fan_b,
                                    y_out, m_out, hebb_out);
}